// IPyramidCapsuleNet_4621384810766
// MI455X (gfx1250) — compile-verified
//
#include <hip/hip_runtime.h>
#include <math.h>

typedef float v2f __attribute__((ext_vector_type(2)));
typedef float v8f __attribute__((ext_vector_type(8)));

#define EPSQ 1e-12f

// ---------------------------------------------------------------- utilities
__global__ void zero_kernel(float* p, int n) {
    int i = blockIdx.x * blockDim.x + threadIdx.x;
    if (i < n) p[i] = 0.f;
}

// 28x28 -> 14x14 average pool (2x2)
__global__ void pool28to14(const float* __restrict__ x, float* __restrict__ x2) {
    int i = blockIdx.x * blockDim.x + threadIdx.x;
    if (i >= 256 * 196) return;
    int b = i / 196, pq = i - b * 196;
    int p = pq / 14, q = pq - p * 14;
    const float* xb = x + (size_t)b * 784;
    x2[i] = 0.25f * (xb[(2 * p) * 28 + 2 * q] + xb[(2 * p) * 28 + 2 * q + 1] +
                     xb[(2 * p + 1) * 28 + 2 * q] + xb[(2 * p + 1) * 28 + 2 * q + 1]);
}

// 14x14 -> 7x7 average pool (2x2)
__global__ void pool14to7(const float* __restrict__ x2, float* __restrict__ x1) {
    int i = blockIdx.x * blockDim.x + threadIdx.x;
    if (i >= 256 * 49) return;
    int b = i / 49, pq = i - b * 49;
    int p = pq / 7, q = pq - p * 7;
    const float* xb = x2 + (size_t)b * 196;
    x1[i] = 0.25f * (xb[(2 * p) * 14 + 2 * q] + xb[(2 * p) * 14 + 2 * q + 1] +
                     xb[(2 * p + 1) * 14 + 2 * q] + xb[(2 * p + 1) * 14 + 2 * q + 1]);
}

// ------------------------------------------------ conv (1->128, stride 1) + ReLU
// output layout: y[c][b][p][q]  (channel-major so BN reductions are contiguous)
template <int KW, int HIN, int HOUT>
__global__ __launch_bounds__(128) void conv_relu(const float* __restrict__ x,   // [256][HIN][HIN]
                                                 const float* __restrict__ w,   // [128][KW][KW]
                                                 const float* __restrict__ bias,
                                                 float* __restrict__ y) {
    __shared__ float patch[KW * KW];
    const int c = threadIdx.x;
    const int idx = blockIdx.x;
    const int b = idx / (HOUT * HOUT);
    const int pq = idx - b * (HOUT * HOUT);
    const int p = pq / HOUT;
    const int q = pq - p * HOUT;
    if (c < KW * KW) {
        int dy = c / KW, dx = c - dy * KW;
        patch[c] = x[(size_t)b * (HIN * HIN) + (p + dy) * HIN + (q + dx)];
    }
    __syncthreads();
    float acc = bias[c];
    const float* wr = w + c * KW * KW;
#pragma unroll
    for (int kk = 0; kk < KW * KW; ++kk) acc = fmaf(patch[kk], wr[kk], acc);
    acc = fmaxf(acc, 0.f);
    y[((size_t)c * 256 + b) * (HOUT * HOUT) + pq] = acc;
}

// ------------------------------------------------ batchnorm stats + apply
__global__ __launch_bounds__(256) void bn_stats(const float* __restrict__ y, int N,
                                                float* __restrict__ mean, float* __restrict__ rstd) {
    __shared__ float s1[256], s2[256];
    const int c = blockIdx.x;
    const float* p = y + (size_t)c * N;
    float a = 0.f, b = 0.f;
    for (int i = threadIdx.x; i < N; i += 256) {
        float v = p[i];
        a += v;
        b += v * v;
    }
    s1[threadIdx.x] = a;
    s2[threadIdx.x] = b;
    __syncthreads();
    for (int st = 128; st > 0; st >>= 1) {
        if (threadIdx.x < st) {
            s1[threadIdx.x] += s1[threadIdx.x + st];
            s2[threadIdx.x] += s2[threadIdx.x + st];
        }
        __syncthreads();
    }
    if (threadIdx.x == 0) {
        float m = s1[0] / (float)N;
        float v = fmaxf(s2[0] / (float)N - m * m, 0.f);
        mean[c] = m;
        rstd[c] = rsqrtf(v + 1e-5f);
    }
}

__global__ void bn_apply(float* __restrict__ y, int N, const float* __restrict__ mean,
                         const float* __restrict__ rstd) {
    int i = blockIdx.x * blockDim.x + threadIdx.x;
    if (i < 128 * N) {
        int c = i / N;
        y[i] = (y[i] - mean[c]) * rstd[c];
    }
}

// ------------------------------------------------ prim conv as implicit GEMM (WMMA fp32)
// C[m (b,s)][c2] = sum_k ynorm * pw ;  M = 256*S rows, N = 256 channels, K = 128*KW*KW
// y layout [cin][b][Hc][Hc]; output layout pout[c2][m] (+ bias).
// Each wave is A-stationary over 4 N-tiles (64 channels): one A fetch feeds 4 WMMAs;
// the 4 B loads use compile-time immediate offsets off one base pointer.
// Block = 8 waves = 2 M-tiles x 4 channel-groups -> 32x256 output tile per block.
template <int KW, int Hc, int S, int W>
__global__ __launch_bounds__(256) void prim_gemm(const float* __restrict__ y,
                                                 const float* __restrict__ pw,  // [256][KTOT]
                                                 const float* __restrict__ pb,  // [256]
                                                 float* __restrict__ pout) {
    constexpr int KK = KW * KW;
    constexpr int KTOT = 128 * KK;
    constexpr int M = 256 * S;
    __shared__ int doff[KTOT];

    const int tid = threadIdx.x;
    for (int kk = tid; kk < KTOT; kk += 256) {
        int cin = kk / KK;
        int rem = kk - cin * KK;
        int dy = rem / KW;
        int dx = rem - dy * KW;
        doff[kk] = cin * (256 * Hc * Hc) + dy * Hc + dx;  // plane stride = B*Hc*Hc
    }
    __syncthreads();

    const int lane = tid & 31;
    const int wave = tid >> 5;   // 0..7
    const int g = lane >> 4;     // 0/1 half-wave
    const int t = lane & 15;
    const int mgrp = wave >> 2;  // 0/1 : which 16-row M tile in this block
    const int wgrp = wave & 3;   // 0..3: which 64-channel group

    const int m_base = blockIdx.x * 32 + mgrp * 16;
    const int c0 = wgrp * 64 + t;  // channel for q-th tile is c0 + q*16

    // A row for this lane: m = m_base + t  ->  (b, oy, ox)
    const int m = m_base + t;
    const int b = m / S;
    const int s = m - b * S;
    const int oy = s / W;
    const int ox = s - oy * W;
    const float* ybase = y + (size_t)b * (Hc * Hc) + (2 * oy) * Hc + 2 * ox;

    const float* wrow = pw + (size_t)c0 * KTOT;

    v8f acc[4] = {{}, {}, {}, {}};
#pragma unroll 2
    for (int k0 = 0; k0 < KTOT; k0 += 4) {
        const int kk = k0 + 2 * g;
        const int o0 = doff[kk];
        const int o1 = doff[kk + 1];
        v2f a;
        a.x = ybase[o0];
        a.y = ybase[o1];
        v2f b0 = *(const v2f*)(wrow + kk);
        v2f b1 = *(const v2f*)(wrow + kk + 16 * KTOT);
        v2f b2 = *(const v2f*)(wrow + kk + 32 * KTOT);
        v2f b3 = *(const v2f*)(wrow + kk + 48 * KTOT);
        acc[0] = __builtin_amdgcn_wmma_f32_16x16x4_f32(false, a, false, b0, (short)0, acc[0], false, false);
        acc[1] = __builtin_amdgcn_wmma_f32_16x16x4_f32(false, a, false, b1, (short)0, acc[1], false, false);
        acc[2] = __builtin_amdgcn_wmma_f32_16x16x4_f32(false, a, false, b2, (short)0, acc[2], false, false);
        acc[3] = __builtin_amdgcn_wmma_f32_16x16x4_f32(false, a, false, b3, (short)0, acc[3], false, false);
    }

    // lane holds C rows (r + 8g), column t -> 8 consecutive m's in [c][m] layout
    float* opb = pout + (size_t)c0 * M + m_base + 8 * g;
#pragma unroll
    for (int q = 0; q < 4; ++q) {
        const float bias = pb[c0 + q * 16];
        float4* op4 = reinterpret_cast<float4*>(opb + (size_t)q * 16 * M);
        op4[0] = make_float4(acc[q][0] + bias, acc[q][1] + bias, acc[q][2] + bias, acc[q][3] + bias);
        op4[1] = make_float4(acc[q][4] + bias, acc[q][5] + bias, acc[q][6] + bias, acc[q][7] + bias);
    }
}

// ------------------------------------------------ capsule routing (one block per (b,n))
template <int R, int S>
__global__ __launch_bounds__(256) void routing_kernel(const float* __restrict__ pfeat, // [256][M]
                                                      const float* __restrict__ rw,    // [10][R][8][16]
                                                      float* __restrict__ logits) {   // [256][10]
    constexpr int M = 256 * S;
    __shared__ float us[R * 8];
    __shared__ float blog[R];
    __shared__ float red2[16 * 256];
    __shared__ float red1[256];
    __shared__ float svv[16];
    __shared__ float vv[16];

    const int tid = threadIdx.x;
    const int b = blockIdx.x;
    const int n = blockIdx.y;

    // u[b,r,:] = p[b, j*32+oc, s], squashed over the 8-dim capsule axis
    for (int r = tid; r < R; r += 256) {
        const int oc = r / S;
        const int s = r - oc * S;
        const int m = b * S + s;
        float u[8];
        float sn = 0.f;
#pragma unroll
        for (int j = 0; j < 8; ++j) {
            u[j] = pfeat[(size_t)(j * 32 + oc) * M + m];
            sn += u[j] * u[j];
        }
        const float sc = sn / (1.f + sn) * rsqrtf(sn + EPSQ);
#pragma unroll
        for (int j = 0; j < 8; ++j) us[r * 8 + j] = u[j] * sc;
    }
    __syncthreads();

    const float* wn = rw + (size_t)n * R * 128;

    auto prior_row = [&](int r, float* acc) {
#pragma unroll
        for (int o = 0; o < 16; ++o) acc[o] = 0.f;
        const float* w = wn + (size_t)r * 128;
#pragma unroll
        for (int j = 0; j < 8; ++j) {
            const float uj = us[r * 8 + j];
#pragma unroll
            for (int o = 0; o < 16; ++o) acc[o] = fmaf(uj, w[j * 16 + o], acc[o]);
        }
    };

    auto col_reduce_and_squash = [&](float* sum16, float invden) {
#pragma unroll
        for (int o = 0; o < 16; ++o) red2[o * 256 + tid] = sum16[o];
        __syncthreads();
        for (int st = 128; st > 0; st >>= 1) {
            if (tid < st) {
#pragma unroll
                for (int o = 0; o < 16; ++o) red2[o * 256 + tid] += red2[o * 256 + tid + st];
            }
            __syncthreads();
        }
        if (tid < 16) svv[tid] = red2[tid * 256] * invden;
        __syncthreads();
        if (tid < 16) {
            float sn = 0.f;
#pragma unroll
            for (int o = 0; o < 16; ++o) sn += svv[o] * svv[o];
            const float sc = sn / (1.f + sn) * rsqrtf(sn + EPSQ);
            vv[tid] = svv[tid] * sc;
        }
        __syncthreads();
    };

    // ---- iteration 0: coupling uniform 1/R
    {
        float sum16[16];
#pragma unroll
        for (int o = 0; o < 16; ++o) sum16[o] = 0.f;
        for (int r = tid; r < R; r += 256) {
            float acc[16];
            prior_row(r, acc);
#pragma unroll
            for (int o = 0; o < 16; ++o) sum16[o] += acc[o];
        }
        col_reduce_and_squash(sum16, 1.f / (float)R);
    }
    // b_logits = <prior_r, v>
    for (int r = tid; r < R; r += 256) {
        float acc[16];
        prior_row(r, acc);
        float d = 0.f;
#pragma unroll
        for (int o = 0; o < 16; ++o) d = fmaf(acc[o], vv[o], d);
        blog[r] = d;
    }
    __syncthreads();

    for (int it = 1; it < 3; ++it) {
        // softmax over routes
        float mloc = -3.402823466e38f;
        for (int r = tid; r < R; r += 256) mloc = fmaxf(mloc, blog[r]);
        red1[tid] = mloc;
        __syncthreads();
        for (int st = 128; st > 0; st >>= 1) {
            if (tid < st) red1[tid] = fmaxf(red1[tid], red1[tid + st]);
            __syncthreads();
        }
        const float mx = red1[0];
        __syncthreads();
        float sloc = 0.f;
        for (int r = tid; r < R; r += 256) sloc += expf(blog[r] - mx);
        red1[tid] = sloc;
        __syncthreads();
        for (int st = 128; st > 0; st >>= 1) {
            if (tid < st) red1[tid] += red1[tid + st];
            __syncthreads();
        }
        const float den = red1[0];
        __syncthreads();

        float sum16[16];
#pragma unroll
        for (int o = 0; o < 16; ++o) sum16[o] = 0.f;
        for (int r = tid; r < R; r += 256) {
            float acc[16];
            prior_row(r, acc);
            const float cw = expf(blog[r] - mx);
#pragma unroll
            for (int o = 0; o < 16; ++o) sum16[o] = fmaf(cw, acc[o], sum16[o]);
        }
        col_reduce_and_squash(sum16, 1.f / den);

        if (it < 2) {
            for (int r = tid; r < R; r += 256) {
                float acc[16];
                prior_row(r, acc);
                float d = 0.f;
#pragma unroll
                for (int o = 0; o < 16; ++o) d = fmaf(acc[o], vv[o], d);
                blog[r] += d;
            }
            __syncthreads();
        }
    }

    if (tid == 0) {
        float sn = 0.f;
#pragma unroll
        for (int o = 0; o < 16; ++o) sn += vv[o] * vv[o];
        atomicAdd(&logits[b * 10 + n], sqrtf(sn + EPSQ));
    }
}

// ------------------------------------------------ final class softmax
__global__ void final_softmax(const float* __restrict__ logits, float* __restrict__ out) {
    int b = blockIdx.x * blockDim.x + threadIdx.x;
    if (b >= 256) return;
    float l[10];
    float mx = -3.402823466e38f;
#pragma unroll
    for (int n = 0; n < 10; ++n) {
        l[n] = logits[b * 10 + n];
        mx = fmaxf(mx, l[n]);
    }
    float sum = 0.f;
#pragma unroll
    for (int n = 0; n < 10; ++n) {
        l[n] = expf(l[n] - mx);
        sum += l[n];
    }
    const float inv = 1.f / sum;
#pragma unroll
    for (int n = 0; n < 10; ++n) out[b * 10 + n] = l[n] * inv;
}

// ================================================================ launcher
extern "C" void kernel_launch(void* const* d_in, const int* in_sizes, int n_in,
                              void* d_out, int out_size, void* d_ws, size_t ws_size,
                              hipStream_t stream) {
    const float* x   = (const float*)d_in[0];
    const float* c1w = (const float*)d_in[1];
    const float* c1b = (const float*)d_in[2];
    const float* p1w = (const float*)d_in[3];
    const float* p1b = (const float*)d_in[4];
    const float* d1w = (const float*)d_in[5];
    const float* c2w = (const float*)d_in[6];
    const float* c2b = (const float*)d_in[7];
    const float* p2w = (const float*)d_in[8];
    const float* p2b = (const float*)d_in[9];
    const float* d2w = (const float*)d_in[10];
    const float* c3w = (const float*)d_in[11];
    const float* c3b = (const float*)d_in[12];
    const float* p3w = (const float*)d_in[13];
    const float* p3b = (const float*)d_in[14];
    const float* d3w = (const float*)d_in[15];

    float* ws = (float*)d_ws;
    float* logits = ws;                    // 2560
    float* x2buf  = ws + 2560;             // 256*196 = 50176
    float* x1buf  = ws + 52736;            // 256*49  = 12544
    float* meanb  = ws + 65280;            // 128
    float* rstdb  = ws + 65408;            // 128
    float* ybuf   = ws + 65536;            // up to 128*256*400 = 13,107,200
    float* pbuf   = ws + 65536 + 13107200; // up to 256*9216   =  2,359,296

    zero_kernel<<<(2560 + 255) / 256, 256, 0, stream>>>(logits, 2560);
    pool28to14<<<(256 * 196 + 255) / 256, 256, 0, stream>>>(x, x2buf);
    pool14to7<<<(256 * 49 + 255) / 256, 256, 0, stream>>>(x2buf, x1buf);

    // ------------- branch 1: 7x7, k=3 -> Hc=5, S=4(W=2), KTOT=1152, M=1024, R=128
    conv_relu<3, 7, 5><<<256 * 25, 128, 0, stream>>>(x1buf, c1w, c1b, ybuf);
    bn_stats<<<128, 256, 0, stream>>>(ybuf, 256 * 25, meanb, rstdb);
    bn_apply<<<(128 * 256 * 25 + 255) / 256, 256, 0, stream>>>(ybuf, 256 * 25, meanb, rstdb);
    prim_gemm<3, 5, 4, 2><<<1024 / 32, 256, 0, stream>>>(ybuf, p1w, p1b, pbuf);
    routing_kernel<128, 4><<<dim3(256, 10), 256, 0, stream>>>(pbuf, d1w, logits);

    // ------------- branch 2: 14x14, k=5 -> Hc=10, S=9(W=3), KTOT=3200, M=2304, R=288
    conv_relu<5, 14, 10><<<256 * 100, 128, 0, stream>>>(x2buf, c2w, c2b, ybuf);
    bn_stats<<<128, 256, 0, stream>>>(ybuf, 256 * 100, meanb, rstdb);
    bn_apply<<<(128 * 256 * 100 + 255) / 256, 256, 0, stream>>>(ybuf, 256 * 100, meanb, rstdb);
    prim_gemm<5, 10, 9, 3><<<2304 / 32, 256, 0, stream>>>(ybuf, p2w, p2b, pbuf);
    routing_kernel<288, 9><<<dim3(256, 10), 256, 0, stream>>>(pbuf, d2w, logits);

    // ------------- branch 3: 28x28, k=9 -> Hc=20, S=36(W=6), KTOT=10368, M=9216, R=1152
    conv_relu<9, 28, 20><<<256 * 400, 128, 0, stream>>>(x, c3w, c3b, ybuf);
    bn_stats<<<128, 256, 0, stream>>>(ybuf, 256 * 400, meanb, rstdb);
    bn_apply<<<(128 * 256 * 400 + 255) / 256, 256, 0, stream>>>(ybuf, 256 * 400, meanb, rstdb);
    prim_gemm<9, 20, 36, 6><<<9216 / 32, 256, 0, stream>>>(ybuf, p3w, p3b, pbuf);
    routing_kernel<1152, 36><<<dim3(256, 10), 256, 0, stream>>>(pbuf, d3w, logits);

    final_softmax<<<1, 256, 0, stream>>>(logits, (float*)d_out);
    (void)in_sizes; (void)n_in; (void)out_size; (void)ws_size;
}